// MultiLevelHierarchicalPrototypes_82995948028556
// MI455X (gfx1250) — compile-verified
//
#include <hip/hip_runtime.h>
#include <hip/hip_bf16.h>

typedef float v2f __attribute__((ext_vector_type(2)));
typedef float v8f __attribute__((ext_vector_type(8)));
typedef int   v4i __attribute__((ext_vector_type(4)));

#if defined(__gfx1250__) && __has_builtin(__builtin_amdgcn_global_load_async_to_lds_b128)
#define HAVE_ASYNC_LDS 1
#else
#define HAVE_ASYNC_LDS 0
#endif

#define EDIM 512
#define NROWS 65536
#define CLS 64
#define CAP 2048
#define MCAP 1280
#define KC 16
#define KCP 20            // LDS k-stride (floats): conflict-free b64 fragment loads

// ---------------------------------------------------------------------------
// 512x512 transpose: Wt[c*512 + k] = W[k*512 + c]
// ---------------------------------------------------------------------------
__global__ __launch_bounds__(256)
void transpose512(const float* __restrict__ W, float* __restrict__ Wt) {
    __shared__ float t[32][33];
    const int bx = blockIdx.x & 15, by = blockIdx.x >> 4;
    const int lx = threadIdx.x & 31, ly = threadIdx.x >> 5;
    #pragma unroll
    for (int i = 0; i < 32; i += 8)
        t[ly + i][lx] = W[(size_t)(by * 32 + ly + i) * EDIM + bx * 32 + lx];
    __syncthreads();
    #pragma unroll
    for (int i = 0; i < 32; i += 8)
        Wt[(size_t)(bx * 32 + ly + i) * EDIM + by * 32 + lx] = t[lx][ly + i];
}

// ---------------------------------------------------------------------------
// GEMM: Y[N,512] = X[N,512] @ W[512,512] + bias, fp32 WMMA 16x16x4.
// W is given PRE-TRANSPOSED (Wt[col][k]) so a B fragment {W[k][c],W[k+1][c]}
// is one ds_load_b64 from LDS. block = 256 threads (8 waves), 32x512 tile;
// wave w: M-tile (w&1), col strip (w>>1)*128 (8 N-tiles, 64 acc VGPRs).
// B fragments for a k-step are prefetched into registers before the WMMA
// burst so DS latency overlaps the multi-cycle WMMAs (dscnt waits > 0).
// ---------------------------------------------------------------------------
__global__ __launch_bounds__(256)
void gemm_bias(const float* __restrict__ X, const float* __restrict__ Wt,
               const float* __restrict__ bias, float* __restrict__ Y) {
    __shared__ float wl[EDIM * KCP];            // 40 KB: wl[col*KCP + kr]

    const int tid   = threadIdx.x;
    const int wv    = tid >> 5;
    const int lane  = tid & 31;
    const int l16   = lane & 15;
    const int hi    = lane >> 4;                // half-wave: K offset 0 or 2
    const int mt    = wv & 1;
    const int nbase = (wv >> 1) * 128;
    const int rowBase = blockIdx.x * 32;

    v8f acc[8];
    #pragma unroll
    for (int t = 0; t < 8; ++t)
        #pragma unroll
        for (int i = 0; i < 8; ++i) acc[t][i] = 0.0f;

    const float* Arow = X + (size_t)(rowBase + mt * 16 + l16) * EDIM;

    for (int kc = 0; kc < EDIM; kc += KC) {
        // stage Wt[:, kc..kc+16) into LDS: 512 cols x 4 float4 each
#if HAVE_ASYNC_LDS
        #pragma unroll
        for (int it = 0; it < 8; ++it) {
            const int f   = it * 256 + tid;     // float4 index 0..2047
            const int col = f >> 2;
            const int j   = f & 3;
            const float* g = Wt + (size_t)col * EDIM + kc + j * 4;
            float* l = &wl[col * KCP + j * 4];
            __builtin_amdgcn_global_load_async_to_lds_b128(
                (v4i*)g, (v4i*)l, 0, 0);
        }
#if __has_builtin(__builtin_amdgcn_s_wait_asynccnt)
        __builtin_amdgcn_s_wait_asynccnt(0);
#else
        asm volatile("s_wait_asynccnt 0x0" ::: "memory");
#endif
#else
        #pragma unroll
        for (int it = 0; it < 8; ++it) {
            const int f   = it * 256 + tid;
            const int col = f >> 2;
            const int j   = f & 3;
            float4 d = *(const float4*)(Wt + (size_t)col * EDIM + kc + j * 4);
            *(float4*)(&wl[col * KCP + j * 4]) = d;
        }
#endif
        __syncthreads();

        #pragma unroll
        for (int ks = 0; ks < KC; ks += 4) {
            const int k  = kc + ks;
            const int kr = ks + hi * 2;
            // prefetch all 8 B fragments for this k-step
            v2f bfr[8];
            #pragma unroll
            for (int t = 0; t < 8; ++t)
                bfr[t] = *(const v2f*)(&wl[(nbase + t * 16 + l16) * KCP + kr]);
            v2f a;
            {   // A frag: lane<16 -> {K,K+1}, lane>=16 -> {K+2,K+3}
                float2 av = *(const float2*)(Arow + k + hi * 2);
                a.x = av.x; a.y = av.y;
            }
            #pragma unroll
            for (int t = 0; t < 8; ++t)
                acc[t] = __builtin_amdgcn_wmma_f32_16x16x4_f32(
                    false, a, false, bfr[t], (short)0, acc[t], false, false);
        }
        __syncthreads();
    }

    // epilogue: C/D vgpr i -> row i (lanes 0-15) / row i+8 (lanes 16-31)
    #pragma unroll
    for (int t = 0; t < 8; ++t) {
        const int col = nbase + t * 16 + l16;
        const float bv = bias[col];
        const int row0 = rowBase + mt * 16 + hi * 8;
        #pragma unroll
        for (int i = 0; i < 8; ++i)
            Y[(size_t)(row0 + i) * EDIM + col] = acc[t][i] + bv;
    }
}

// ---------------------------------------------------------------------------
// Row-wise LayerNorm + ReLU, in place. One wave per row (8 rows / block).
// ---------------------------------------------------------------------------
__global__ __launch_bounds__(256)
void ln_relu(float* __restrict__ Xio, const float* __restrict__ gamma,
             const float* __restrict__ beta) {
    const int wv = threadIdx.x >> 5, lane = threadIdx.x & 31;
    const size_t row = (size_t)blockIdx.x * 8 + wv;
    float* p = Xio + row * EDIM;

    float x[16], s = 0.f, s2 = 0.f;
    #pragma unroll
    for (int i = 0; i < 16; ++i) {
        x[i] = p[lane + i * 32];
        s += x[i]; s2 += x[i] * x[i];
    }
    #pragma unroll
    for (int off = 16; off; off >>= 1) {
        s  += __shfl_xor(s,  off, 32);
        s2 += __shfl_xor(s2, off, 32);
    }
    const float mu  = s * (1.0f / 512.0f);
    const float var = s2 * (1.0f / 512.0f) - mu * mu;
    const float inv = rsqrtf(var + 1e-5f);
    #pragma unroll
    for (int i = 0; i < 16; ++i) {
        const int e = lane + i * 32;
        float y = (x[i] - mu) * inv * gamma[e] + beta[e];
        p[e] = fmaxf(y, 0.0f);
    }
}

// ---------------------------------------------------------------------------
// Deterministic per-class member lists (ballot + prefix compaction, no atomics)
// ---------------------------------------------------------------------------
__global__ __launch_bounds__(256)
void build_lists(const int* __restrict__ labels, int* __restrict__ counts,
                 int* __restrict__ members, int n) {
    __shared__ int wsum[8];
    const int c = blockIdx.x;
    const int tid = threadIdx.x, wv = tid >> 5, lane = tid & 31;
    int base = 0;
    for (int n0 = 0; n0 < n; n0 += 256) {
        const int idx = n0 + tid;
        const bool flag = (labels[idx] == c);
        const unsigned mask = (unsigned)__ballot(flag);
        const int prefix = __builtin_popcount(mask & ((1u << lane) - 1u));
        if (lane == 0) wsum[wv] = __builtin_popcount(mask);
        __syncthreads();
        int wbase = 0;
        for (int i = 0; i < wv; ++i) wbase += wsum[i];
        int tot = 0;
        for (int i = 0; i < 8; ++i) tot += wsum[i];
        if (flag) members[c * CAP + base + wbase + prefix] = idx;
        base += tot;
        __syncthreads();
    }
    if (tid == 0) counts[c] = base;
}

// ---------------------------------------------------------------------------
// ctx[c,:] = mean of F rows belonging to class c
// ---------------------------------------------------------------------------
__global__ __launch_bounds__(256)
void class_mean(const float* __restrict__ F, const int* __restrict__ counts,
                const int* __restrict__ members, float* __restrict__ ctx) {
    const int c = blockIdx.x, tid = threadIdx.x;
    const int cnt = counts[c];
    float a0 = 0.f, a1 = 0.f;
    for (int m = 0; m < cnt; ++m) {
        const int r = members[c * CAP + m];
        const float* p = F + (size_t)r * EDIM;
        a0 += p[tid]; a1 += p[tid + 256];
    }
    const float inv = 1.0f / (float)cnt;
    ctx[c * EDIM + tid]       = a0 * inv;
    ctx[c * EDIM + tid + 256] = a1 * inv;
}

// ---------------------------------------------------------------------------
// q = ctx @ Wq + bq   ([64,512] x [512,512], tiny)
// ---------------------------------------------------------------------------
__global__ __launch_bounds__(256)
void qproj(const float* __restrict__ ctx, const float* __restrict__ Wq,
           const float* __restrict__ bq, float* __restrict__ q) {
    __shared__ float cs[EDIM];
    const int c = blockIdx.x, tid = threadIdx.x;
    cs[tid] = ctx[c * EDIM + tid];
    cs[tid + 256] = ctx[c * EDIM + tid + 256];
    __syncthreads();
    float a0 = bq[tid], a1 = bq[tid + 256];
    for (int j = 0; j < EDIM; ++j) {
        const float cv = cs[j];
        a0 += cv * Wq[(size_t)j * EDIM + tid];
        a1 += cv * Wq[(size_t)j * EDIM + tid + 256];
    }
    q[c * EDIM + tid] = a0;
    q[c * EDIM + tid + 256] = a1;
}

// ---------------------------------------------------------------------------
// Masked per-class multi-head attention. One block per class, one wave per
// head for scores/softmax; scores kept in LDS.
// ---------------------------------------------------------------------------
__global__ __launch_bounds__(256)
void attention(const float* __restrict__ q, const float* __restrict__ K,
               const float* __restrict__ V, const int* __restrict__ counts,
               const int* __restrict__ members, float* __restrict__ outbuf) {
    __shared__ float qs[EDIM];
    __shared__ float sc[8 * MCAP];
    const int c = blockIdx.x, tid = threadIdx.x, wv = tid >> 5, lane = tid & 31;
    int cnt = counts[c]; if (cnt > MCAP) cnt = MCAP;

    qs[tid] = q[c * EDIM + tid];
    qs[tid + 256] = q[c * EDIM + tid + 256];
    __syncthreads();

    const float scale = 0.125f;                  // 1/sqrt(64)
    for (int m = lane; m < cnt; m += 32) {       // head = wv
        const int r = members[c * CAP + m];
        const float* kp = K + (size_t)r * EDIM + wv * 64;
        float d = 0.f;
        #pragma unroll
        for (int j = 0; j < 64; ++j) d += qs[wv * 64 + j] * kp[j];
        sc[wv * MCAP + m] = d * scale;
    }
    __syncthreads();

    float mx = -3.4e38f;
    for (int m = lane; m < cnt; m += 32) mx = fmaxf(mx, sc[wv * MCAP + m]);
    #pragma unroll
    for (int off = 16; off; off >>= 1) mx = fmaxf(mx, __shfl_xor(mx, off, 32));
    float sum = 0.f;
    for (int m = lane; m < cnt; m += 32) {
        const float e = expf(sc[wv * MCAP + m] - mx);
        sc[wv * MCAP + m] = e;
        sum += e;
    }
    #pragma unroll
    for (int off = 16; off; off >>= 1) sum += __shfl_xor(sum, off, 32);
    const float inv = 1.0f / sum;
    for (int m = lane; m < cnt; m += 32) sc[wv * MCAP + m] *= inv;
    __syncthreads();

    const int h0 = tid >> 6;
    const int h1 = (tid + 256) >> 6;
    float o0 = 0.f, o1 = 0.f;
    for (int m = 0; m < cnt; ++m) {
        const int r = members[c * CAP + m];
        const float* vp = V + (size_t)r * EDIM;
        o0 += sc[h0 * MCAP + m] * vp[tid];
        o1 += sc[h1 * MCAP + m] * vp[tid + 256];
    }
    outbuf[c * EDIM + tid] = o0;
    outbuf[c * EDIM + tid + 256] = o1;
}

// ---------------------------------------------------------------------------
// final += softmax(level_weights)[l]/temps[l] * (attnout @ Wo + bo)
// ---------------------------------------------------------------------------
__global__ __launch_bounds__(256)
void proto_accum(const float* __restrict__ ob, const float* __restrict__ Wo,
                 const float* __restrict__ bo, const float* __restrict__ lvlw,
                 const float* __restrict__ lvlt, float* __restrict__ out, int l) {
    __shared__ float os[EDIM];
    const int c = blockIdx.x, tid = threadIdx.x;
    os[tid] = ob[c * EDIM + tid];
    os[tid + 256] = ob[c * EDIM + tid + 256];
    __syncthreads();

    const float w0 = lvlw[0], w1 = lvlw[1], w2 = lvlw[2];
    const float mx = fmaxf(w0, fmaxf(w1, w2));
    const float e0 = expf(w0 - mx), e1 = expf(w1 - mx), e2 = expf(w2 - mx);
    const float lw = (l == 0 ? e0 : l == 1 ? e1 : e2) / (e0 + e1 + e2);
    const float coef = lw / lvlt[l];

    float a0 = bo[tid], a1 = bo[tid + 256];
    for (int j = 0; j < EDIM; ++j) {
        const float v = os[j];
        a0 += v * Wo[(size_t)j * EDIM + tid];
        a1 += v * Wo[(size_t)j * EDIM + tid + 256];
    }
    if (l == 0) {
        out[c * EDIM + tid] = coef * a0;
        out[c * EDIM + tid + 256] = coef * a1;
    } else {
        out[c * EDIM + tid] += coef * a0;
        out[c * EDIM + tid + 256] += coef * a1;
    }
}

// ---------------------------------------------------------------------------
extern "C" void kernel_launch(void* const* d_in, const int* in_sizes, int n_in,
                              void* d_out, int out_size, void* d_ws, size_t ws_size,
                              hipStream_t stream) {
    const float* X    = (const float*)d_in[0];
    const int*   lab  = (const int*)d_in[1];
    const float* W1   = (const float*)d_in[2];
    const float* b1   = (const float*)d_in[3];
    const float* gam  = (const float*)d_in[4];
    const float* bet  = (const float*)d_in[5];
    const float* W2   = (const float*)d_in[6];
    const float* b2   = (const float*)d_in[7];
    const float* Wq   = (const float*)d_in[8];
    const float* bq   = (const float*)d_in[9];
    const float* Wk   = (const float*)d_in[10];
    const float* bk   = (const float*)d_in[11];
    const float* Wv   = (const float*)d_in[12];
    const float* bv   = (const float*)d_in[13];
    const float* Wo   = (const float*)d_in[14];
    const float* bo   = (const float*)d_in[15];
    const float* lvlw = (const float*)d_in[16];
    const float* lvlt = (const float*)d_in[17];
    float* out = (float*)d_out;

    char* ws = (char*)d_ws;
    size_t off = 0;
    int* counts  = (int*)(ws + off);   off += 256;
    int* members = (int*)(ws + off);   off += (size_t)CLS * CAP * 4;
    float* ctx   = (float*)(ws + off); off += (size_t)CLS * EDIM * 4;
    float* qb    = (float*)(ws + off); off += (size_t)CLS * EDIM * 4;
    float* ob    = (float*)(ws + off); off += (size_t)CLS * EDIM * 4;
    const size_t wsz = (size_t)EDIM * EDIM * 4;        // 1 MB per weight
    float* W1t = (float*)(ws + off); off += wsz;
    float* W2t = (float*)(ws + off); off += wsz;
    float* Wkt = (float*)(ws + off); off += wsz;
    float* Wvt = (float*)(ws + off); off += wsz;
    const size_t big = (size_t)NROWS * EDIM * 4;
    float* B0 = (float*)(ws + off); off += big;   // T -> H -> K
    float* B1 = (float*)(ws + off); off += big;   // F
    float* B2 = (float*)(ws + off); off += big;   // V

    build_lists<<<CLS, 256, 0, stream>>>(lab, counts, members, NROWS);
    transpose512<<<256, 256, 0, stream>>>(Wk, Wkt);
    transpose512<<<256, 256, 0, stream>>>(Wv, Wvt);

    for (int l = 0; l < 3; ++l) {
        const float* W1l = W1 + (size_t)l * EDIM * EDIM;
        const float* W2l = W2 + (size_t)l * EDIM * EDIM;
        transpose512<<<256, 256, 0, stream>>>(W1l, W1t);
        transpose512<<<256, 256, 0, stream>>>(W2l, W2t);
        gemm_bias<<<NROWS / 32, 256, 0, stream>>>(X,  W1t, b1 + l * EDIM, B0);
        ln_relu  <<<NROWS / 8,  256, 0, stream>>>(B0, gam + l * EDIM, bet + l * EDIM);
        gemm_bias<<<NROWS / 32, 256, 0, stream>>>(B0, W2t, b2 + l * EDIM, B1);   // F
        gemm_bias<<<NROWS / 32, 256, 0, stream>>>(B1, Wkt, bk, B0);              // K
        gemm_bias<<<NROWS / 32, 256, 0, stream>>>(B1, Wvt, bv, B2);              // V
        class_mean<<<CLS, 256, 0, stream>>>(B1, counts, members, ctx);
        qproj     <<<CLS, 256, 0, stream>>>(ctx, Wq, bq, qb);
        attention <<<CLS, 256, 0, stream>>>(qb, B0, B2, counts, members, ob);
        proto_accum<<<CLS, 256, 0, stream>>>(ob, Wo, bo, lvlw, lvlt, out, l);
    }
}